// GaussianBlurConv_48687749267714
// MI455X (gfx1250) — compile-verified
//
#include <hip/hip_runtime.h>
#include <math.h>

typedef float v2f __attribute__((ext_vector_type(2)));
typedef float v8f __attribute__((ext_vector_type(8)));

#define IMGS   96          // 16 batch * 6 channels
#define NDIM   512
#define PADW   552         // 512 + 2*20
#define KTAPS  41
#define LDS_STRIDE 556     // padded row stride in LDS (even, bank-spreading)
#define STRIP  16
#define NCHUNK 14          // ceil(56 / 4) K-steps of the 16x16x4 WMMA

// Horizontal 41-tap blur along the fastest axis with reflect padding,
// output stored TRANSPOSED. Two applications == separable 2-D blur with
// reflect padding on both axes, layout restored.
__global__ __launch_bounds__(256)
void GaussianBlurConv_hblur_t(const float* __restrict__ in,
                              const float* __restrict__ w2d,
                              float* __restrict__ outT)
{
    __shared__ float smem[STRIP * LDS_STRIDE + 64];
    float* taps = &smem[STRIP * LDS_STRIDE];

    const int tid   = threadIdx.x;
    const int strip = blockIdx.x;              // 0..31
    const int img   = blockIdx.y;              // 0..95
    const int row0  = strip * STRIP;
    const size_t imgBase = (size_t)img * NDIM * NDIM;

    // 1) recover the exact separable 1-D profile from the 2-D kernel:
    //    h = u u^T  =>  u[i] = h[i][20] / sqrt(h[20][20])
    if (tid < KTAPS) {
        float center = w2d[20 * KTAPS + 20];
        taps[tid] = w2d[tid * KTAPS + 20] / sqrtf(center);
    }

    // 2) stage 16 rows x 552 reflect-padded columns into LDS
    for (int i = tid; i < STRIP * PADW; i += 256) {
        int r = i / PADW;
        int p = i - r * PADW;
        int c = p - 20;                        // padded -> source column
        if (c < 0)        c = -c;              // reflect (no edge repeat)
        if (c > NDIM - 1) c = 2 * (NDIM - 1) - c;
        smem[r * LDS_STRIDE + p] = in[imgBase + (size_t)(row0 + r) * NDIM + c];
    }
    __syncthreads();

    // 3) per-lane Toeplitz weight band B[j][n] = u[j - n], j in 4k + {0..3}
    const int lane = tid & 31;
    const int wv   = tid >> 5;                 // wave id 0..7
    const int hi   = lane >> 4;                // lane half
    const int nn   = lane & 15;                // N / M-row index
    v2f bw[NCHUNK];
#pragma unroll
    for (int k = 0; k < NCHUNK; ++k) {
        int i0 = 4 * k + 2 * hi - nn;          // B VGPR0: K rows {0,2}
        int i1 = i0 + 1;                       // B VGPR1: K rows {1,3}
        bw[k].x = (i0 >= 0 && i0 < KTAPS) ? taps[i0] : 0.0f;
        bw[k].y = (i1 >= 0 && i1 < KTAPS) ? taps[i1] : 0.0f;
    }

    // 4) each wave computes 4 16x16 output tiles: D = A(16x56) * Toeplitz(56x16)
    const float* arow = &smem[nn * LDS_STRIDE + 2 * hi]; // A row for this lane
#pragma unroll 1
    for (int tt = 0; tt < 4; ++tt) {
        const int tbase = (wv * 4 + tt) * 16;  // first output column of tile
        v8f c = {};
#pragma unroll
        for (int k = 0; k < NCHUNK; ++k) {
            v2f a = *(const v2f*)(arow + tbase + 4 * k);
            c = __builtin_amdgcn_wmma_f32_16x16x4_f32(
                    false, a, false, bw[k], (short)0, c, false, false);
        }
        // transposed store: outT[img][x0+n][row0+m], m = v + 8*hi contiguous
        float* dst = &outT[imgBase + (size_t)(tbase + nn) * NDIM + row0 + 8 * hi];
        reinterpret_cast<float4*>(dst)[0] = make_float4(c[0], c[1], c[2], c[3]);
        reinterpret_cast<float4*>(dst)[1] = make_float4(c[4], c[5], c[6], c[7]);
    }
}

extern "C" void kernel_launch(void* const* d_in, const int* in_sizes, int n_in,
                              void* d_out, int out_size, void* d_ws, size_t ws_size,
                              hipStream_t stream) {
    (void)in_sizes; (void)n_in; (void)out_size; (void)ws_size;
    const float* x   = (const float*)d_in[0];   // [16,6,512,512]
    const float* w2d = (const float*)d_in[1];   // [6,1,41,41] (same kernel per ch)
    float* out = (float*)d_out;
    float* tmp = (float*)d_ws;                  // 96*512*512 floats (~100 MB)

    dim3 grid(NDIM / STRIP, IMGS);              // (32, 96)
    // pass 1: blur x, store transposed into workspace
    GaussianBlurConv_hblur_t<<<grid, 256, 0, stream>>>(x, w2d, tmp);
    // pass 2: blur (former) y axis, transpose back into d_out
    GaussianBlurConv_hblur_t<<<grid, 256, 0, stream>>>(tmp, w2d, out);
}